// LearnerModel_15728170238460
// MI455X (gfx1250) — compile-verified
//
#include <hip/hip_runtime.h>
#include <math.h>

#define HID   2048
#define VOCAB 32000
#define MAXLEN 128

typedef __attribute__((ext_vector_type(2))) float v2f;
typedef __attribute__((ext_vector_type(8))) float v8f;

// ---------------------------------------------------------------------------
// Wave-level GEMV tile via V_WMMA_F32_16X16X4_F32.
// Computes y[row0+m] += sum_k W[row0+m][k] * x[k] for m = 0..15 (one wave).
// A = 16x4 tile of W (ISA layout: lanes 0-15 hold K=0,1; lanes 16-31 K=2,3).
// B = x[k..k+3] broadcast across all 16 columns, so every column of D is y.
// ---------------------------------------------------------------------------
__device__ __forceinline__ void wave_gemv_wmma(const float* __restrict__ W,
                                               long ldw, long row0,
                                               const float* __restrict__ x,
                                               int K, v8f& acc) {
  const int lane = threadIdx.x & 31;
  const int m    = lane & 15;
  const int kh   = (lane >> 4) * 2;          // 0 for lanes 0-15, 2 for 16-31
  const float* __restrict__ wrow = W + (row0 + m) * ldw;
#pragma unroll 8
  for (int k = 0; k < K; k += 4) {
    v2f a, b;
    a.x = wrow[k + kh];                      // contiguous 8B -> global_load_b64
    a.y = wrow[k + kh + 1];
    b.x = x[k + kh];                         // broadcast (hits WGP$)
    b.y = x[k + kh + 1];
    acc = __builtin_amdgcn_wmma_f32_16x16x4_f32(
        /*neg_a=*/false, a, /*neg_b=*/false, b,
        /*c_mod=*/(short)0, acc, /*reuse_a=*/false, /*reuse_b=*/false);
  }
}

// D layout (32-bit 16x16): VGPR j, lanes 0-15 -> M=j; lanes 16-31 -> M=8+j.
// With broadcast-B every column is identical, so lanes 0 and 16 hold y[0..7]
// and y[8..15] respectively in their 8 accumulator VGPRs.
#define ACT_NONE 0
#define ACT_TANH 1
#define ACT_RELU 2

__device__ __forceinline__ void wave_store16(float* __restrict__ y, long row0,
                                             const v8f& acc,
                                             const float* __restrict__ bias,
                                             const float* __restrict__ bias2,
                                             int act) {
  const int lane = threadIdx.x & 31;
  if ((lane & 15) == 0) {                    // lanes 0 and 16
    const int off = (lane >> 4) * 8;
#pragma unroll
    for (int j = 0; j < 8; ++j) {
      long r = row0 + off + j;
      float v = acc[j];
      if (bias)  v += bias[r];
      if (bias2) v += bias2[r];
      if (act == ACT_TANH) v = tanhf(v);
      else if (act == ACT_RELU) v = fmaxf(v, 0.0f);
      y[r] = v;
    }
  }
}

// ---------------------------------------------------------------------------
// Embedding gather: x = emb[tok]; written into low halves of both concat bufs.
// ---------------------------------------------------------------------------
__global__ void prep_kernel(const int* __restrict__ tok,
                            const float* __restrict__ emb,
                            float* __restrict__ cat1lo,
                            float* __restrict__ cat2lo) {
  int i = blockIdx.x * blockDim.x + threadIdx.x;
  if (i < HID) {
    float v = emb[(long)tok[0] * HID + i];
    cat1lo[i] = v;
    cat2lo[i] = v;
  }
}

// ---------------------------------------------------------------------------
// Elman cell: hout = tanh(W_ih x + b_ih + W_hh h + b_hh). 16 blocks x 8 waves.
// ---------------------------------------------------------------------------
__global__ __launch_bounds__(256) void rnn_cell_kernel(
    const float* __restrict__ Wih, const float* __restrict__ Whh,
    const float* __restrict__ bih, const float* __restrict__ bhh,
    const float* __restrict__ x,   const float* __restrict__ h,
    float* __restrict__ hout) {
  long tile = (long)blockIdx.x * 8 + (threadIdx.x >> 5);
  long row0 = tile * 16;
  v8f acc = {};
  wave_gemv_wmma(Wih, HID, row0, x, HID, acc);
  wave_gemv_wmma(Whh, HID, row0, h, HID, acc);
  wave_store16(hout, row0, acc, bih, bhh, ACT_TANH);
}

// ---------------------------------------------------------------------------
// Generic GEMV: y = act(W x + b). One wave per 16 rows.
// ---------------------------------------------------------------------------
__global__ __launch_bounds__(256) void gemv_kernel(
    const float* __restrict__ W, const float* __restrict__ bias,
    const float* __restrict__ x, float* __restrict__ y, int K, int act) {
  long tile = (long)blockIdx.x * 8 + (threadIdx.x >> 5);
  long row0 = tile * 16;
  v8f acc = {};
  wave_gemv_wmma(W, K, row0, x, K, acc);
  wave_store16(y, row0, acc, bias, nullptr, act);
}

// ---------------------------------------------------------------------------
// Attention: scores = attn_W @ cat1 + attn_b (8 WMMA tiles, one per wave),
// softmax over 128, context = w @ encoder_outputs -> cat2 high half.
// Single block of 256 threads (tiny: ~3 MB traffic total).
// ---------------------------------------------------------------------------
__global__ __launch_bounds__(256) void attn_kernel(
    const float* __restrict__ attn_W, const float* __restrict__ attn_b,
    const float* __restrict__ cat1,   const float* __restrict__ enc,
    float* __restrict__ cat2hi) {
  __shared__ float s_sc[MAXLEN];
  __shared__ float s_w[MAXLEN];
  const int wave = threadIdx.x >> 5;
  const int lane = threadIdx.x & 31;
  long row0 = (long)wave * 16;

  v8f acc = {};
  wave_gemv_wmma(attn_W, 2 * HID, row0, cat1, 2 * HID, acc);
  if ((lane & 15) == 0) {
    const int off = (lane >> 4) * 8;
#pragma unroll
    for (int j = 0; j < 8; ++j)
      s_sc[row0 + off + j] = acc[j] + attn_b[row0 + off + j];
  }
  __syncthreads();

  if (threadIdx.x == 0) {                    // 128 elems: serial is negligible
    float mx = s_sc[0];
    for (int l = 1; l < MAXLEN; ++l) mx = fmaxf(mx, s_sc[l]);
    float sum = 0.0f;
    for (int l = 0; l < MAXLEN; ++l) { float e = expf(s_sc[l] - mx); s_w[l] = e; sum += e; }
    float inv = 1.0f / sum;
    for (int l = 0; l < MAXLEN; ++l) s_w[l] *= inv;
  }
  __syncthreads();

  for (int c = threadIdx.x; c < HID; c += 256) {   // coalesced over columns
    float s = 0.0f;
    for (int l = 0; l < MAXLEN; ++l) s += s_w[l] * enc[(long)l * HID + c];
    cat2hi[c] = s;
  }
}

// ---------------------------------------------------------------------------
// log_softmax over 32000 logits: one block, shared-memory tree reductions.
// ---------------------------------------------------------------------------
__global__ __launch_bounds__(1024) void logsoftmax_kernel(
    const float* __restrict__ logits, float* __restrict__ out, int n) {
  __shared__ float red[1024];
  const int tid = threadIdx.x;

  float mx = -INFINITY;
  for (int i = tid; i < n; i += 1024) mx = fmaxf(mx, logits[i]);
  red[tid] = mx; __syncthreads();
  for (int s = 512; s > 0; s >>= 1) {
    if (tid < s) red[tid] = fmaxf(red[tid], red[tid + s]);
    __syncthreads();
  }
  mx = red[0]; __syncthreads();

  float sum = 0.0f;
  for (int i = tid; i < n; i += 1024) sum += expf(logits[i] - mx);
  red[tid] = sum; __syncthreads();
  for (int s = 512; s > 0; s >>= 1) {
    if (tid < s) red[tid] += red[tid + s];
    __syncthreads();
  }
  float lse = mx + logf(red[0]);

  for (int i = tid; i < n; i += 1024) out[i] = logits[i] - lse;
}

// ---------------------------------------------------------------------------
// Launch
// ---------------------------------------------------------------------------
extern "C" void kernel_launch(void* const* d_in, const int* in_sizes, int n_in,
                              void* d_out, int out_size, void* d_ws, size_t ws_size,
                              hipStream_t stream) {
  const int*   tok    = (const int*)  d_in[0];
  const float* hidden = (const float*)d_in[1];
  const float* enc    = (const float*)d_in[2];
  const float* emb    = (const float*)d_in[3];
  const float* attn_W = (const float*)d_in[4];
  const float* attn_b = (const float*)d_in[5];
  const float* comb_W = (const float*)d_in[6];
  const float* comb_b = (const float*)d_in[7];
  const float* W_ih   = (const float*)d_in[8];
  const float* W_hh   = (const float*)d_in[9];
  const float* b_ih   = (const float*)d_in[10];
  const float* b_hh   = (const float*)d_in[11];
  const float* out_W  = (const float*)d_in[12];
  const float* out_b  = (const float*)d_in[13];

  // Workspace layout (floats)
  float* cat1   = (float*)d_ws;          // [x | h1]          : 4096
  float* cat2   = cat1 + 2 * HID;        // [x | attn_applied]: 4096
  float* o      = cat2 + 2 * HID;        // relu(comb)        : 2048
  float* outh   = o + HID;               // rnn step 2        : 2048
  float* logits = outh + HID;            // pre-softmax       : 32000

  // 1) x = emb[tok] into both concat buffers' low halves
  prep_kernel<<<(HID + 255) / 256, 256, 0, stream>>>(tok, emb, cat1, cat2);

  // 2) h1 = tanh(W_ih x + b_ih + W_hh h0 + b_hh) -> cat1 high half
  rnn_cell_kernel<<<HID / 16 / 8, 256, 0, stream>>>(
      W_ih, W_hh, b_ih, b_hh, cat1, hidden, cat1 + HID);

  // 3) attention: scores over cat1, softmax, context -> cat2 high half
  attn_kernel<<<1, 256, 0, stream>>>(attn_W, attn_b, cat1, enc, cat2 + HID);

  // 4) o = relu(comb_W cat2 + comb_b)
  gemv_kernel<<<HID / 16 / 8, 256, 0, stream>>>(comb_W, comb_b, cat2, o,
                                                2 * HID, ACT_RELU);

  // 5) out_h = tanh(W_ih o + b_ih + W_hh h1 + b_hh)
  rnn_cell_kernel<<<HID / 16 / 8, 256, 0, stream>>>(
      W_ih, W_hh, b_ih, b_hh, o, cat1 + HID, outh);

  // 6) logits = out_W out_h + out_b   (262 MB stream -> the bandwidth term)
  gemv_kernel<<<VOCAB / 16 / 8, 256, 0, stream>>>(out_W, out_b, outh, logits,
                                                  HID, ACT_NONE);

  // 7) log_softmax -> d_out
  logsoftmax_kernel<<<1, 1024, 0, stream>>>(logits, (float*)d_out, VOCAB);
}